// MultiEffectFusion_42176578847125
// MI455X (gfx1250) — compile-verified
//
#include <hip/hip_runtime.h>
#include <math.h>

// ---------------- geometry ----------------
#define T16 16
#define NN 400
#define FF 64
#define DIM 6400          // T16*NN
#define TPT 25            // tiles per t (400/16)

typedef __attribute__((ext_vector_type(2))) float v2f;
typedef __attribute__((ext_vector_type(8))) float v8f;

// 16x16 fp32 Gram tile: C = A(16x64) * B(16x64)^T via 16x V_WMMA_F32_16X16X4_F32.
// A-frag layout (16x4): lanes 0-15 -> M=lane, K=kk+0/kk+1 ; lanes 16-31 -> M=lane-16, K=kk+2/kk+3.
// B-frag (4x16) mirrors with N, so a Gram matrix feeds identical per-lane data to A and B.
__device__ __forceinline__ v8f gram16x16(const float* __restrict__ Arows,
                                         const float* __restrict__ Brows,
                                         int lda, int ldb, int lane) {
  const int rl = lane & 15;
  const int hi = lane >> 4;         // 0: K pair 0/1, 1: K pair 2/3
  v8f c = {0.f, 0.f, 0.f, 0.f, 0.f, 0.f, 0.f, 0.f};
#pragma unroll
  for (int kk = 0; kk < FF; kk += 4) {
    const int ka = kk + hi * 2;
    v2f a, b;
    a.x = Arows[rl * lda + ka];
    a.y = Arows[rl * lda + ka + 1];
    b.x = Brows[rl * ldb + ka];
    b.y = Brows[rl * ldb + ka + 1];
    c = __builtin_amdgcn_wmma_f32_16x16x4_f32(false, a, false, b, (short)0, c,
                                              false, false);
  }
  return c;
}

__device__ __forceinline__ float sigf(float v) {
  return 1.f / (1.f + __expf(-v));
}

// ---------------- kernel 1: GRU over time (per node) ----------------
// grid=400 blocks (one per node), block=64 threads (one per hidden unit)
__global__ void gru_kernel(const float* __restrict__ x,    // [T,N,F]
                           const float* __restrict__ Wih,  // [192,64]
                           const float* __restrict__ Whh,  // [192,64]
                           const float* __restrict__ bih,  // [192]
                           const float* __restrict__ bhh,  // [192]
                           float* __restrict__ temp) {     // [T,N,H]
  __shared__ float sWih[192 * 65];  // stride 65 -> bank-conflict-free
  __shared__ float sWhh[192 * 65];
  __shared__ float xs[64];
  __shared__ float hs[64];
  const int n = blockIdx.x;
  const int j = threadIdx.x;  // hidden index

  for (int i = j; i < 192 * 64; i += 64) {
    const int r = i >> 6, k = i & 63;
    sWih[r * 65 + k] = Wih[i];
    sWhh[r * 65 + k] = Whh[i];
  }
  const float bir = bih[j],      biz = bih[64 + j],  bin_ = bih[128 + j];
  const float bhr = bhh[j],      bhz = bhh[64 + j],  bhn = bhh[128 + j];
  float h = 0.f;
  hs[j] = 0.f;
  __syncthreads();

  const float* wr = &sWih[j * 65];
  const float* wz = &sWih[(64 + j) * 65];
  const float* wn = &sWih[(128 + j) * 65];
  const float* ur = &sWhh[j * 65];
  const float* uz = &sWhh[(64 + j) * 65];
  const float* un = &sWhh[(128 + j) * 65];

  for (int t = 0; t < T16; ++t) {
    xs[j] = x[(t * NN + n) * FF + j];
    __syncthreads();
    float gir = bir, giz = biz, gin = bin_;
    float ghr = bhr, ghz = bhz, ghn = bhn;
#pragma unroll 8
    for (int k = 0; k < 64; ++k) {
      const float xv = xs[k], hv = hs[k];
      gir += wr[k] * xv;  giz += wz[k] * xv;  gin += wn[k] * xv;
      ghr += ur[k] * hv;  ghz += uz[k] * hv;  ghn += un[k] * hv;
    }
    const float r = sigf(gir + ghr);
    const float z = sigf(giz + ghz);
    const float nn = tanhf(gin + r * ghn);
    h = (1.f - z) * nn + z * h;
    __syncthreads();
    hs[j] = h;
    temp[(t * NN + n) * FF + j] = h;
    __syncthreads();
  }
}

// ---------------- kernel 2: masked attention -> hid [T,N,F] ----------------
// grid=(7,16), block=64: one thread per row i of time-slice t; x[t] staged in LDS.
__global__ void attn_kernel(const float* __restrict__ x,
                            const int* __restrict__ adj,
                            float* __restrict__ hid) {
  __shared__ float xs[NN * FF];  // 102.4 KB (<=320KB per WGP)
  const int t = blockIdx.y;
  const int i = blockIdx.x * 64 + threadIdx.x;
  const float* xt = x + (size_t)t * NN * FF;
  for (int idx = threadIdx.x; idx < NN * FF / 4; idx += 64)
    ((float4*)xs)[idx] = ((const float4*)xt)[idx];
  __syncthreads();
  if (i >= NN) return;

  float xi[FF];
#pragma unroll
  for (int k = 0; k < FF; ++k) xi[k] = xs[i * FF + k];
  const int* arow = adj + i * NN;

  // pass 1: online max/sum
  float m = -3.0e38f, Z = 0.f;
  for (int j = 0; j < NN; ++j) {
    float s = 0.f;
#pragma unroll
    for (int k = 0; k < FF; ++k) s += xi[k] * xs[j * FF + k];
    s *= 0.125f;
    if (arow[j] == 0) s = -1e12f;
    if (s > m) { Z *= __expf(m - s); m = s; }
    Z += __expf(s - m);
  }
  const float invZ = 1.f / Z;

  // pass 2: weighted accumulate
  float acc[FF];
#pragma unroll
  for (int k = 0; k < FF; ++k) acc[k] = 0.f;
  for (int j = 0; j < NN; ++j) {
    float s = 0.f;
#pragma unroll
    for (int k = 0; k < FF; ++k) s += xi[k] * xs[j * FF + k];
    s *= 0.125f;
    if (arow[j] == 0) s = -1e12f;
    const float w = __expf(s - m) * invZ;
#pragma unroll
    for (int k = 0; k < FF; ++k) acc[k] += w * xs[j * FF + k];
  }
  float* o = hid + ((size_t)t * NN + i) * FF;
#pragma unroll
  for (int k = 0; k < FF; ++k) o[k] = acc[k];
}

// ---------------- kernel 3: tw[t] = sigmoid(temp[t]@temp[t]^T / 8) ----------------
// one wave per 16x16 tile; 10000 tiles; block=256 (8 waves), grid=1250.
__global__ void tw_kernel(const float* __restrict__ temp,
                          float* __restrict__ tw) {
  const int wave = (blockIdx.x * blockDim.x + threadIdx.x) >> 5;
  const int lane = threadIdx.x & 31;
  const int t = wave / (TPT * TPT);
  const int rem = wave % (TPT * TPT);
  const int rb = rem / TPT, cb = rem % TPT;
  const float* A = temp + ((size_t)t * NN + rb * 16) * FF;
  const float* B = temp + ((size_t)t * NN + cb * 16) * FF;
  v8f c = gram16x16(A, B, FF, FF, lane);
  const int col = cb * 16 + (lane & 15);
  const int rbase = rb * 16 + ((lane >> 4) << 3);
  float* o = tw + (size_t)t * NN * NN;
#pragma unroll
  for (int v = 0; v < 8; ++v)
    o[(rbase + v) * NN + col] = sigf(0.125f * c[v]);
}

// ---------------- kernel 4: sw[n] = sigmoid(sp[n]@sp[n]^T / 8) ----------------
// one wave per node; sp[n] rows strided by N*F in hid. block=128, grid=100.
__global__ void sw_kernel(const float* __restrict__ hid,
                          float* __restrict__ sw) {
  const int n = (blockIdx.x * blockDim.x + threadIdx.x) >> 5;
  const int lane = threadIdx.x & 31;
  const float* S = hid + (size_t)n * FF;  // row t at offset t*N*F
  v8f c = gram16x16(S, S, NN * FF, NN * FF, lane);
  const int col = lane & 15;
  const int rbase = (lane >> 4) << 3;
  float* o = sw + (size_t)n * 256;
#pragma unroll
  for (int v = 0; v < 8; ++v)
    o[(rbase + v) * 16 + col] = sigf(0.125f * c[v]);
}

// ---------------- kernel 5: fused direct-GEMM + gate softmax ----------------
// one wave per 16x16 output tile; grid=(100 col-tile groups, 400 row tiles), block=128.
__global__ void fuse_kernel(const float* __restrict__ x,
                            const int* __restrict__ adj,
                            const float* __restrict__ tw,
                            const float* __restrict__ sw,
                            const float* __restrict__ pTW,
                            const float* __restrict__ pSW,
                            const float* __restrict__ pDW,
                            float* __restrict__ out) {
  const int lane = threadIdx.x & 31;
  const int ct = blockIdx.x * 4 + (threadIdx.x >> 5);  // col tile 0..399
  const int rt = blockIdx.y;                           // row tile 0..399
  const int t = rt / TPT, nb = (rt % TPT) * 16;
  const int tp = ct / TPT, mb = (ct % TPT) * 16;
  const float gTW = pTW[0], gSW = pSW[0], gDW = pDW[0];

  const float* A = x + ((size_t)t * NN + nb) * FF;
  const float* B = x + ((size_t)tp * NN + mb) * FF;
  v8f c = gram16x16(A, B, FF, FF, lane);  // direct-effect tile

  const int col = lane & 15;
  const int nprime = mb + col;
  const int rbase = (lane >> 4) << 3;
  const float* twp = tw + (size_t)tp * NN * NN;
  const size_t orow0 = (size_t)(rt * 16 + rbase) * DIM + (size_t)ct * 16 + col;

#pragma unroll
  for (int v = 0; v < 8; ++v) {
    const int n = nb + rbase + v;  // row node
    const float f = (adj[n * NN + nprime] != 0) ? 1.f : 0.f;
    const float a = twp[n * NN + nprime] * f;          // time:  tw[t', n, n']
    const float b = sw[n * 256 + tp * 16 + t] * f;     // space: sw[n, t', t]
    const float d = sigf(0.125f * c[v]) * f;           // direct
    const float e0 = __expf(gTW * a);
    const float e1 = __expf(gSW * b);
    const float e2 = __expf(gDW * d);
    out[orow0 + (size_t)v * DIM] = (e0 * a + e1 * b + e2 * d) / (e0 + e1 + e2);
  }
}

// ---------------- launch ----------------
extern "C" void kernel_launch(void* const* d_in, const int* in_sizes, int n_in,
                              void* d_out, int out_size, void* d_ws,
                              size_t ws_size, hipStream_t stream) {
  (void)in_sizes; (void)n_in; (void)out_size; (void)ws_size;
  const float* x   = (const float*)d_in[0];  // [T,N,F]
  const int*   adj = (const int*)d_in[1];    // [N,N]
  const float* Wih = (const float*)d_in[2];
  const float* Whh = (const float*)d_in[3];
  const float* bih = (const float*)d_in[4];
  const float* bhh = (const float*)d_in[5];
  const float* TW  = (const float*)d_in[6];
  const float* SW  = (const float*)d_in[7];
  const float* DW  = (const float*)d_in[8];
  float* out = (float*)d_out;

  char* ws = (char*)d_ws;
  float* temp = (float*)(ws);                               // 1,638,400 B
  float* tw   = (float*)(ws + 1638400);                     // 10,240,000 B
  float* hid  = (float*)(ws + 1638400 + 10240000);          // 1,638,400 B
  float* sw   = (float*)(ws + 1638400 + 10240000 + 1638400);//   409,600 B

  gru_kernel<<<NN, 64, 0, stream>>>(x, Wih, Whh, bih, bhh, temp);
  attn_kernel<<<dim3(7, T16), 64, 0, stream>>>(x, adj, hid);
  tw_kernel<<<1250, 256, 0, stream>>>(temp, tw);     // 10000 waves
  sw_kernel<<<100, 128, 0, stream>>>(hid, sw);       // 400 waves
  fuse_kernel<<<dim3(100, NN), 128, 0, stream>>>(x, adj, tw, sw, TW, SW, DW, out);
}